// GCN_78683800863474
// MI455X (gfx1250) — compile-verified
//
#include <hip/hip_runtime.h>
#include <hip/hip_bf16.h>

// ---------------------------------------------------------------------------
// 3-layer GCN for MI455X (gfx1250, wave32).
//   GEMM:    V_WMMA_F32_16X16X4_F32 (exact fp32, matrix pipe)
//   Scatter: native global_atomic_add_f32, L2-resident working set (<192MB)
// ---------------------------------------------------------------------------

typedef float v2f __attribute__((ext_vector_type(2)));
typedef float v8f __attribute__((ext_vector_type(8)));

#define D 256  // feature width (fixed by reference)

__device__ __forceinline__ void atomic_add_f32(float* p, float v) {
#if defined(__AMDGCN__)
    unsafeAtomicAdd(p, v);   // -> global_atomic_add_f32 (no CAS loop)
#else
    atomicAdd(p, v);
#endif
}

// ---- degree / normalization prep ------------------------------------------

__global__ void k_fill1(float* __restrict__ deg, int n) {
    int i = blockIdx.x * blockDim.x + threadIdx.x;
    if (i < n) deg[i] = 1.0f;               // self-loop contributes 1
}

__global__ void k_degree(const int* __restrict__ dst, float* __restrict__ deg, int E) {
    int e = blockIdx.x * blockDim.x + threadIdx.x;
    if (e < E) atomic_add_f32(&deg[dst[e]], 1.0f);
}

__global__ void k_rsqrt_inplace(float* __restrict__ deg, int n) {
    int i = blockIdx.x * blockDim.x + threadIdx.x;
    if (i < n) deg[i] = rsqrtf(deg[i]);     // deg >= 1 always (self-loop)
}

__global__ void k_norm(const int* __restrict__ src, const int* __restrict__ dst,
                       const float* __restrict__ dinv, float* __restrict__ norm, int E) {
    int e = blockIdx.x * blockDim.x + threadIdx.x;
    if (e < E) norm[e] = dinv[src[e]] * dinv[dst[e]];
}

// W[k][n] (row-major [in,out]) -> Wt[n][k] so B fragments are contiguous.
__global__ void k_transpose256(const float* __restrict__ W, float* __restrict__ Wt) {
    int k = blockIdx.x;      // 256 blocks
    int n = threadIdx.x;     // 256 threads
    Wt[(size_t)n * D + k] = W[(size_t)k * D + n];
}

// ---- fp32 WMMA GEMM: T = H @ W  (H: [N,256], Wt: [256,256] col-major) -----
// Block = 128 threads (4 waves). Block owns a 16-row strip; wave w owns
// column tiles 4w..4w+3. Per K-step(4): 1 A-frag load (reused 4x) + 4 B-frag
// loads + 4 v_wmma_f32_16x16x4_f32.
__global__ __launch_bounds__(128) void k_gemm_wmma(const float* __restrict__ H,
                                                   const float* __restrict__ Wt,
                                                   float* __restrict__ T) {
    const int lane = threadIdx.x & 31;
    const int wave = threadIdx.x >> 5;       // 0..3
    const int half = lane >> 4;              // K sub-block select (ISA layout)
    const int ln   = lane & 15;
    const int r0   = blockIdx.x << 4;        // 16-row strip (N % 16 == 0)

    // A fragment: lane(ln,half) holds row r0+ln, K = {k+2*half, k+2*half+1}
    const float* aptr = H + (size_t)(r0 + ln) * D + (half << 1);

    const float* bptr[4];
    v8f acc[4] = {};
#pragma unroll
    for (int c = 0; c < 4; ++c) {
        const int col = (((wave << 2) + c) << 4) + ln;   // B col this lane owns
        bptr[c] = Wt + (size_t)col * D + (half << 1);
    }

    for (int k = 0; k < D; k += 4) {
        v2f a = *(const v2f*)(aptr + k);
#pragma unroll
        for (int c = 0; c < 4; ++c) {
            v2f b = *(const v2f*)(bptr[c] + k);
            acc[c] = __builtin_amdgcn_wmma_f32_16x16x4_f32(
                false, a, false, b, (short)0, acc[c], false, false);
        }
    }

    // D layout: vgpr i / lane(ln,half) -> row r0 + 8*half + i, col tile + ln
#pragma unroll
    for (int c = 0; c < 4; ++c) {
        const int col = (((wave << 2) + c) << 4) + ln;
#pragma unroll
        for (int i = 0; i < 8; ++i) {
            const int row = r0 + (half << 3) + i;
            T[(size_t)row * D + col] = acc[c][i];
        }
    }
}

// ---- edge scatter: agg[dst] += norm * T[src] ------------------------------
// One 256-thread block per edge: coalesced 1KB row gather + per-cacheline
// grouped f32 atomics (t and agg are both L2-resident on MI455X: 192MB L2).
__global__ __launch_bounds__(256) void k_scatter(const float* __restrict__ T,
                                                 const int* __restrict__ src,
                                                 const int* __restrict__ dst,
                                                 const float* __restrict__ norm,
                                                 float* __restrict__ agg) {
    const int e = blockIdx.x;
    const int j = threadIdx.x;
    const int s = src[e];       // uniform per block -> scalar loads
    const int d = dst[e];
    const float w = norm[e];
    const float v = T[(size_t)s * D + j] * w;
    atomic_add_f32(&agg[(size_t)d * D + j], v);
}

// ---- epilogue: self-loop (no atomics needed) + bias (+ReLU), in place -----
__global__ __launch_bounds__(256) void k_finish(float* __restrict__ agg,
                                                const float* __restrict__ T,
                                                const float* __restrict__ dinv,
                                                const float* __restrict__ bias,
                                                int relu) {
    const int i = blockIdx.x;
    const int j = threadIdx.x;
    const size_t idx = (size_t)i * D + j;
    const float di = dinv[i];
    float v = agg[idx] + T[idx] * di * di + bias[j];
    if (relu) v = fmaxf(v, 0.0f);
    agg[idx] = v;
}

// ---------------------------------------------------------------------------

extern "C" void kernel_launch(void* const* d_in, const int* in_sizes, int n_in,
                              void* d_out, int out_size, void* d_ws, size_t ws_size,
                              hipStream_t stream) {
    const float* x   = (const float*)d_in[0];
    const int*   ei  = (const int*)d_in[1];
    const float* W1  = (const float*)d_in[2];
    const float* b1  = (const float*)d_in[3];
    const float* W2  = (const float*)d_in[4];
    const float* b2  = (const float*)d_in[5];
    const float* W3  = (const float*)d_in[6];
    const float* b3  = (const float*)d_in[7];
    float* out = (float*)d_out;

    const int N = in_sizes[0] / D;     // 50000
    const int E = in_sizes[1] / 2;     // 800000
    const int* srcI = ei;
    const int* dstI = ei + E;

    // workspace carve-up (floats): dinv | norm | 3x Wt | t | agg  (~107 MB)
    float* ws   = (float*)d_ws;
    float* dinv = ws;
    float* norm = dinv + ((N + 63) & ~63);
    float* Wt   = norm + ((E + 63) & ~63);
    float* t    = Wt + 3 * D * D;
    float* agg  = t + (size_t)N * D;

    const int TPB = 256;
    const size_t featBytes = (size_t)N * D * sizeof(float);

    // ---- normalization prep ----
    k_fill1<<<(N + TPB - 1) / TPB, TPB, 0, stream>>>(dinv, N);
    k_degree<<<(E + TPB - 1) / TPB, TPB, 0, stream>>>(dstI, dinv, E);
    k_rsqrt_inplace<<<(N + TPB - 1) / TPB, TPB, 0, stream>>>(dinv, N);
    k_norm<<<(E + TPB - 1) / TPB, TPB, 0, stream>>>(srcI, dstI, dinv, norm, E);
    k_transpose256<<<D, D, 0, stream>>>(W1, Wt + 0 * D * D);
    k_transpose256<<<D, D, 0, stream>>>(W2, Wt + 1 * D * D);
    k_transpose256<<<D, D, 0, stream>>>(W3, Wt + 2 * D * D);

    const int gemmGrid = N / 16;       // 3125, exact

    // ---- layer 1: x -> agg (in-place h1) ----
    k_gemm_wmma<<<gemmGrid, 128, 0, stream>>>(x, Wt + 0 * D * D, t);
    hipMemsetAsync(agg, 0, featBytes, stream);
    k_scatter<<<E, TPB, 0, stream>>>(t, srcI, dstI, norm, agg);
    k_finish<<<N, TPB, 0, stream>>>(agg, t, dinv, b1, 1);

    // ---- layer 2: agg -> agg (gemm reads agg before it is re-zeroed) ----
    k_gemm_wmma<<<gemmGrid, 128, 0, stream>>>(agg, Wt + 1 * D * D, t);
    hipMemsetAsync(agg, 0, featBytes, stream);
    k_scatter<<<E, TPB, 0, stream>>>(t, srcI, dstI, norm, agg);
    k_finish<<<N, TPB, 0, stream>>>(agg, t, dinv, b2, 1);

    // ---- layer 3: agg -> d_out (no ReLU) ----
    k_gemm_wmma<<<gemmGrid, 128, 0, stream>>>(agg, Wt + 2 * D * D, t);
    hipMemsetAsync(out, 0, featBytes, stream);
    k_scatter<<<E, TPB, 0, stream>>>(t, srcI, dstI, norm, out);
    k_finish<<<N, TPB, 0, stream>>>(out, t, dinv, b3, 0);
}